// MAPR_31018253812178
// MI455X (gfx1250) — compile-verified
//
#include <hip/hip_runtime.h>

#define NN 100000
#define MM 1600000
#define DIM 128
#define ALPHA 0.1f
#define GAMMA 0.9f

typedef float v2f __attribute__((ext_vector_type(2)));
typedef float v8f __attribute__((ext_vector_type(8)));

// ---------------------------------------------------------------------------
// Fused FP32 WMMA dual-GEMM: Cv = A @ Bv, Cs = A @ Bs (A shared, loaded once).
// One wave per 16x16 output tile pair; V_WMMA_F32_16X16X4_F32, K stepped by 4
// (2 x 32 = 64 wmma per wave, two independent accumulator chains).
// A 16x4 layout:  lane L -> m=L%16, holds K = kb+2*(L/16) .. +1  (2 VGPRs)
// B 4x16 layout:  lane L -> n=L%16, VGPR r holds K = kb+2*(L/16)+r
// C 16x16 layout: lane L -> n=L%16, VGPR r holds m = r + 8*(L/16)
// ---------------------------------------------------------------------------
__global__ void __launch_bounds__(32)
gemm_wmma_dual(const float* __restrict__ A,
               const float* __restrict__ Bv, const float* __restrict__ Bs,
               float* __restrict__ Cv, float* __restrict__ Cs) {
  const int tm   = blockIdx.x;          // tile along M (rows of A)
  const int tn   = blockIdx.y;          // tile along N (cols of B), 0..7
  const int lane = threadIdx.x;         // 0..31, wave32
  const int lo   = lane & 15;
  const int half = lane >> 4;           // 0 or 1

  const int m = tm * 16 + lo;           // A row for this lane
  const int n = tn * 16 + lo;           // B/C column for this lane

  v8f accv = {};
  v8f accs = {};
#pragma unroll
  for (int kb = 0; kb < DIM; kb += 4) {
    const int ka = kb + 2 * half;
    v2f a = *(const v2f*)(A + (size_t)m * DIM + ka);   // 8B aligned (ka even)
    v2f bv, bs;
    bv.x = Bv[(size_t)ka * DIM + n];
    bv.y = Bv[(size_t)(ka + 1) * DIM + n];
    bs.x = Bs[(size_t)ka * DIM + n];
    bs.y = Bs[(size_t)(ka + 1) * DIM + n];
    accv = __builtin_amdgcn_wmma_f32_16x16x4_f32(
        false, a, false, bv, (short)0, accv, false, false);
    accs = __builtin_amdgcn_wmma_f32_16x16x4_f32(
        false, a, false, bs, (short)0, accs, false, false);
  }
#pragma unroll
  for (int r = 0; r < 8; ++r) {
    const int row = tm * 16 + r + 8 * half;
    Cv[(size_t)row * DIM + n] = accv[r];
    Cs[(size_t)row * DIM + n] = accs[r];
  }
}

// ---------------------------------------------------------------------------
// CSR build helpers
// ---------------------------------------------------------------------------
__global__ void zero_int_kernel(int* __restrict__ p, int n) {
  for (int i = blockIdx.x * blockDim.x + threadIdx.x; i < n;
       i += gridDim.x * blockDim.x)
    p[i] = 0;
}

__global__ void degree_kernel(const int* __restrict__ e, int* __restrict__ deg) {
  for (int m = blockIdx.x * blockDim.x + threadIdx.x; m < MM;
       m += gridDim.x * blockDim.x)
    atomicAdd(&deg[e[m]], 1);           // e[0..M) = destination (e0)
}

// Single-block exclusive scan: rowptr[0]=0, rowptr[i+1]=sum deg[0..i]
__global__ void __launch_bounds__(1024)
scan_kernel(const int* __restrict__ deg, int* __restrict__ rowptr, int n) {
  __shared__ int s[1024];
  __shared__ int carry;
  const int tid = threadIdx.x;
  if (tid == 0) { carry = 0; rowptr[0] = 0; }
  __syncthreads();
  for (int base = 0; base < n; base += 1024) {
    const int i = base + tid;
    s[tid] = (i < n) ? deg[i] : 0;
    __syncthreads();
    for (int off = 1; off < 1024; off <<= 1) {
      int t = (tid >= off) ? s[tid - off] : 0;
      __syncthreads();
      s[tid] += t;
      __syncthreads();
    }
    if (i < n) rowptr[i + 1] = carry + s[tid];
    __syncthreads();
    if (tid == 0) carry += s[1023];
    __syncthreads();
  }
}

__global__ void copy_cursor_kernel(const int* __restrict__ rowptr,
                                   int* __restrict__ cursor, int n) {
  for (int i = blockIdx.x * blockDim.x + threadIdx.x; i < n;
       i += gridDim.x * blockDim.x)
    cursor[i] = rowptr[i];
}

__global__ void fill_kernel(const int* __restrict__ e, int* __restrict__ cursor,
                            int* __restrict__ col) {
  for (int m = blockIdx.x * blockDim.x + threadIdx.x; m < MM;
       m += gridDim.x * blockDim.x) {
    const int d   = e[m];               // destination = e0
    const int src = e[MM + m];          // source      = e1
    const int pos = atomicAdd(&cursor[d], 1);
    col[pos] = src;
  }
}

// ---------------------------------------------------------------------------
// z0 = ALPHA * v   (first power-iteration step collapses since z starts at 0)
// ---------------------------------------------------------------------------
__global__ void zinit_kernel(const float* __restrict__ v, float* __restrict__ z) {
  const int n4 = NN * DIM / 4;
  const float4* vp = (const float4*)v;
  float4* zp = (float4*)z;
  for (int i = blockIdx.x * blockDim.x + threadIdx.x; i < n4;
       i += gridDim.x * blockDim.x) {
    float4 t = vp[i];
    zp[i] = make_float4(ALPHA * t.x, ALPHA * t.y, ALPHA * t.z, ALPHA * t.w);
  }
}

// ---------------------------------------------------------------------------
// Pull-based SpMV sweep: one wave32 per node; lane owns a float4 column slice.
// z_out[d] = GAMMA * (1/(deg+eps)) * sum_{s in N(d)} z_in[s]  +  ALPHA * v[d]
// No atomics; gathers are L2-resident (z = 51.2 MB << 192 MB L2).
// ---------------------------------------------------------------------------
__global__ void __launch_bounds__(256)
spmv_kernel(const int* __restrict__ rowptr, const int* __restrict__ col,
            const float* __restrict__ zin, const float* __restrict__ v,
            float* __restrict__ zout) {
  const int row  = (blockIdx.x * blockDim.x + threadIdx.x) >> 5;
  const int lane = threadIdx.x & 31;
  if (row >= NN) return;
  const int beg = rowptr[row];
  const int end = rowptr[row + 1];
  const float4* Z = (const float4*)zin;
  float ax = 0.f, ay = 0.f, az = 0.f, aw = 0.f;
  for (int m = beg; m < end; ++m) {
    const int s = col[m];
    if (m + 8 < end)
      __builtin_prefetch(zin + (size_t)col[m + 8] * DIM + lane * 4, 0, 3);
    float4 t = Z[(size_t)s * 32 + lane];
    ax += t.x; ay += t.y; az += t.z; aw += t.w;
  }
  const float w   = 1.0f / ((float)(end - beg) + 1e-16f);
  const float gw  = GAMMA * w;
  float4 vv = ((const float4*)v)[(size_t)row * 32 + lane];
  float4 o;
  o.x = gw * ax + ALPHA * vv.x;
  o.y = gw * ay + ALPHA * vv.y;
  o.z = gw * az + ALPHA * vv.z;
  o.w = gw * aw + ALPHA * vv.w;
  ((float4*)zout)[(size_t)row * 32 + lane] = o;
}

// ---------------------------------------------------------------------------
// out = LayerNorm(z + skip + lin_b) * ln_g + ln_b ; skip lives in d_out and is
// overwritten in place (each thread reads its own elements before writing).
// One wave per row; wave reduction via __shfl_xor (wave32).
// ---------------------------------------------------------------------------
__global__ void __launch_bounds__(256)
norm_kernel(const float* __restrict__ z, const float* __restrict__ lin_b,
            const float* __restrict__ ln_g, const float* __restrict__ ln_b,
            float* __restrict__ out /* holds skip on entry */) {
  const int row  = (blockIdx.x * blockDim.x + threadIdx.x) >> 5;
  const int lane = threadIdx.x & 31;
  if (row >= NN) return;
  float4 zz = ((const float4*)z)[(size_t)row * 32 + lane];
  float4 sk = ((const float4*)out)[(size_t)row * 32 + lane];
  float4 lb = ((const float4*)lin_b)[lane];
  zz.x += sk.x + lb.x; zz.y += sk.y + lb.y;
  zz.z += sk.z + lb.z; zz.w += sk.w + lb.w;

  float s1 = zz.x + zz.y + zz.z + zz.w;
  float s2 = zz.x * zz.x + zz.y * zz.y + zz.z * zz.z + zz.w * zz.w;
#pragma unroll
  for (int off = 16; off > 0; off >>= 1) {
    s1 += __shfl_xor(s1, off, 32);
    s2 += __shfl_xor(s2, off, 32);
  }
  const float mu   = s1 * (1.0f / 128.0f);
  const float var  = s2 * (1.0f / 128.0f) - mu * mu;
  const float rinv = rsqrtf(var + 1e-5f);
  float4 g = ((const float4*)ln_g)[lane];
  float4 b = ((const float4*)ln_b)[lane];
  float4 o;
  o.x = (zz.x - mu) * rinv * g.x + b.x;
  o.y = (zz.y - mu) * rinv * g.y + b.y;
  o.z = (zz.z - mu) * rinv * g.z + b.z;
  o.w = (zz.w - mu) * rinv * g.w + b.w;
  ((float4*)out)[(size_t)row * 32 + lane] = o;
}

// ---------------------------------------------------------------------------
extern "C" void kernel_launch(void* const* d_in, const int* in_sizes, int n_in,
                              void* d_out, int out_size, void* d_ws, size_t ws_size,
                              hipStream_t stream) {
  const float* x      = (const float*)d_in[0];
  const int*   e      = (const int*)d_in[1];   // [0..M)=e0 (dst), [M..2M)=e1 (src)
  const float* lin_w  = (const float*)d_in[2];
  const float* lin_b  = (const float*)d_in[3];
  const float* skip_w = (const float*)d_in[4];
  const float* ln_g   = (const float*)d_in[5];
  const float* ln_b   = (const float*)d_in[6];
  float* out = (float*)d_out;

  // workspace layout
  float* v  = (float*)d_ws;                 // N*D
  float* z0 = v  + (size_t)NN * DIM;        // N*D
  float* z1 = z0 + (size_t)NN * DIM;        // N*D
  int* deg    = (int*)(z1 + (size_t)NN * DIM);
  int* rowptr = deg + NN;                   // N+1
  int* cursor = rowptr + NN + 1;            // N
  int* col    = cursor + NN;                // M

  // 1) Fused dual GEMM via fp32 WMMA: v = x@lin_w ; skip = x@skip_w -> d_out
  //    (x read once; two independent WMMA accumulator chains per wave)
  dim3 ggrid(NN / 16, DIM / 16);
  gemm_wmma_dual<<<ggrid, 32, 0, stream>>>(x, lin_w, skip_w, v, out);

  // 2) CSR build (grouped by destination e0)
  zero_int_kernel<<<512, 256, 0, stream>>>(deg, NN);
  degree_kernel<<<2048, 256, 0, stream>>>(e, deg);
  scan_kernel<<<1, 1024, 0, stream>>>(deg, rowptr, NN);
  copy_cursor_kernel<<<512, 256, 0, stream>>>(rowptr, cursor, NN);
  fill_kernel<<<2048, 256, 0, stream>>>(e, cursor, col);

  // 3) Power iteration: z = ALPHA*v, then 9 pull-SpMV sweeps (ping-pong)
  zinit_kernel<<<2048, 256, 0, stream>>>(v, z0);
  float* cur = z0;
  float* nxt = z1;
  const int spmv_grid = (NN * 32 + 255) / 256;  // one wave per node
  for (int it = 0; it < 9; ++it) {
    spmv_kernel<<<spmv_grid, 256, 0, stream>>>(rowptr, col, cur, v, nxt);
    float* t = cur; cur = nxt; nxt = t;
  }

  // 4) skip + bias + LayerNorm (in place over d_out)
  norm_kernel<<<spmv_grid, 256, 0, stream>>>(cur, lin_b, ln_g, ln_b, out);
}